// BroadcastEdgeUpdate_18597208392249
// MI455X (gfx1250) — compile-verified
//
#include <hip/hip_runtime.h>
#include <hip/hip_bf16.h>

typedef float v2f __attribute__((ext_vector_type(2)));
typedef float v8f __attribute__((ext_vector_type(8)));

#define LN_EPS 1e-5f

constexpr int CZ   = 128;   // c_z
constexpr int CA   = 16;    // c_atompair
constexpr int NRES = 512;   // n_res

constexpr int ROWS_PER_WG = 64;    // rows of length 128 handled per workgroup
constexpr int THREADS1    = 128;   // 4 waves (wave32)
constexpr int LDS_STRIDE  = 132;   // 128 + 4 pad: bank = (4*row + col) % 64 -> conflict-free A reads

// ---------------------------------------------------------------------------
// Kernel 1: fused LayerNorm + Linear(128 -> 16, no bias) via V_WMMA_F32_16X16X4_F32
//   z: [n_rows, 128] fp32 (flattened [nb*nr*nr, cz])
//   U: [n_rows, 16]  fp32 scratch (flatish_update)
// ---------------------------------------------------------------------------
__global__ __launch_bounds__(THREADS1)
void ln_linear_wmma_kernel(const float* __restrict__ z,
                           const float* __restrict__ gamma,
                           const float* __restrict__ beta,
                           const float* __restrict__ W,   // [16, 128] row-major
                           float* __restrict__ U)
{
    __shared__ float zn[ROWS_PER_WG][LDS_STRIDE];

    const int t         = threadIdx.x;
    const int lane      = t & 31;
    const int wave      = t >> 5;
    const int wgRowBase = blockIdx.x * ROWS_PER_WG;

    // ---- Phase 1: LayerNorm 64 rows into LDS (2 threads per row) ----------
    {
        const int row  = t >> 1;     // 0..63
        const int half = t & 1;      // which 64-float half of the row
        const float4* src = (const float4*)(z + (size_t)(wgRowBase + row) * CZ + half * 64);

        float4 x[16];
        float s = 0.f, ss = 0.f;
        #pragma unroll
        for (int i = 0; i < 16; ++i) {
            x[i] = src[i];
            s  += x[i].x + x[i].y + x[i].z + x[i].w;
            ss += x[i].x * x[i].x + x[i].y * x[i].y + x[i].z * x[i].z + x[i].w * x[i].w;
        }
        // partner (t ^ 1) lives in the same wave32 -> shuffle reduce
        s  += __shfl_xor(s,  1, 32);
        ss += __shfl_xor(ss, 1, 32);

        const float mu   = s * (1.0f / (float)CZ);
        const float var  = ss * (1.0f / (float)CZ) - mu * mu;
        const float rstd = rsqrtf(var + LN_EPS);

        const float4* g4 = (const float4*)(gamma + half * 64);
        const float4* b4 = (const float4*)(beta  + half * 64);
        #pragma unroll
        for (int i = 0; i < 16; ++i) {
            const float4 g = g4[i], b = b4[i], v = x[i];
            float* d = &zn[row][half * 64 + i * 4];
            d[0] = (v.x - mu) * rstd * g.x + b.x;
            d[1] = (v.y - mu) * rstd * g.y + b.y;
            d[2] = (v.z - mu) * rstd * g.z + b.z;
            d[3] = (v.w - mu) * rstd * g.w + b.w;
        }
    }

    // ---- Phase 0b: preload B fragments (W transposed) into registers ------
    // B[k][n] = W[n][k]. ISA B layout for 16x16x4 f32:
    //   VGPR0: lanes 0-15 = B[4k+0][n=lane], lanes 16-31 = B[4k+2][n=lane-16]
    //   VGPR1: lanes 0-15 = B[4k+1][n],      lanes 16-31 = B[4k+3][n]
    // => lane l holds float2 W[l&15][4k + 2*(l>>4) + {0,1}]  (8B aligned)
    const int bn   = lane & 15;
    const int boff = (lane >> 4) * 2;
    v2f bfrag[32];
    #pragma unroll
    for (int kk = 0; kk < 32; ++kk) {
        const float* wp = W + (size_t)bn * CZ + 4 * kk + boff;
        bfrag[kk] = v2f{wp[0], wp[1]};
    }

    __syncthreads();

    // ---- Phase 2: 16x16 output tile per wave, K=128 as 32 x (16x16x4) -----
    // ISA A layout: lane l -> M = l&15, K-pair = 4k + 2*(l>>4) + {0,1}
    const int rA    = wave * 16 + (lane & 15);
    const int cAoff = (lane >> 4) * 2;

    v8f acc = {};  // C = 0
    #pragma unroll
    for (int kk = 0; kk < 32; ++kk) {
        const float* ap = &zn[rA][4 * kk + cAoff];   // 8B-aligned -> ds_load_b64
        v2f a = v2f{ap[0], ap[1]};
        acc = __builtin_amdgcn_wmma_f32_16x16x4_f32(
            /*neg_a=*/false, a, /*neg_b=*/false, bfrag[kk],
            /*c_mod=*/(short)0, acc, /*reuse_a=*/false, /*reuse_b=*/false);
    }

    // ---- Phase 3: store D. VGPR j: lanes 0-15 -> M=j, lanes 16-31 -> M=j+8
    const int outRowBase = wgRowBase + wave * 16 + ((lane >> 4) << 3);
    const int nCol       = lane & 15;
    #pragma unroll
    for (int j = 0; j < 8; ++j) {
        U[(size_t)(outRowBase + j) * CA + nCol] = acc[j];
    }
}

// ---------------------------------------------------------------------------
// Kernel 2: per-edge double-indirection gather of 64B rows from U (L2-resident)
// ---------------------------------------------------------------------------
__global__ __launch_bounds__(256)
void edge_gather_kernel(const int* __restrict__ edge_index,   // [2, n_edges]
                        const int* __restrict__ atom_res,     // [n_atoms]
                        const float* __restrict__ U,          // [nb*nres*nres, 16]
                        float* __restrict__ out,              // [n_edges, 16]
                        int n_edges)
{
    const int e = blockIdx.x * blockDim.x + threadIdx.x;
    if (e >= n_edges) return;

    const int a0 = edge_index[e];
    const int a1 = edge_index[n_edges + e];
    const int r0 = atom_res[a0];            // flat residue index in [0, nb*nres)
    const int r1 = atom_res[a1] % NRES;     // second index mod n_res

    const float4* src = (const float4*)(U + ((size_t)r0 * NRES + r1) * CA);
    float4* dst = (float4*)(out + (size_t)e * CA);
    dst[0] = src[0];
    dst[1] = src[1];
    dst[2] = src[2];
    dst[3] = src[3];
}

extern "C" void kernel_launch(void* const* d_in, const int* in_sizes, int n_in,
                              void* d_out, int out_size, void* d_ws, size_t ws_size,
                              hipStream_t stream) {
    const float* z        = (const float*)d_in[0];
    const float* gamma    = (const float*)d_in[1];
    const float* beta     = (const float*)d_in[2];
    const float* W        = (const float*)d_in[3];
    const int*   atom_res = (const int*)d_in[4];
    const int*   edges    = (const int*)d_in[5];
    float* out = (float*)d_out;
    float* U   = (float*)d_ws;   // needs n_rows * 16 * 4 = 16.8 MB scratch

    const int n_rows  = in_sizes[0] / CZ;   // 262144 = nb * nres * nres
    const int n_edges = in_sizes[5] / 2;    // 1,000,000

    ln_linear_wmma_kernel<<<n_rows / ROWS_PER_WG, THREADS1, 0, stream>>>(
        z, gamma, beta, W, U);

    edge_gather_kernel<<<(n_edges + 255) / 256, 256, 0, stream>>>(
        edges, atom_res, U, out, n_edges);
}